// FOFE_Encoding_41996190220715
// MI455X (gfx1250) — compile-verified
//
#include <hip/hip_runtime.h>
#include <hip/hip_bf16.h>

#define NUM_WORDS       262144
#define WORD_LEN        32
#define VOCAB           256
#define WAVES_PER_BLOCK 8
#define WORDS_PER_WAVE  4
#define WORDS_PER_BLOCK (WAVES_PER_BLOCK * WORDS_PER_WAVE)   // 32
#define BLOCK_THREADS   (WAVES_PER_BLOCK * 32)               // 256 = 8 wave32

typedef __attribute__((ext_vector_type(4))) float v4f;
typedef int v4i __attribute__((vector_size(16)));

// Pointer types matching the async-store builtin's expected parameters:
//   param0: global (addrspace 1) int4*   param1: LDS (addrspace 3) int4*
typedef __attribute__((address_space(1))) v4i* gv4i_ptr;
typedef __attribute__((address_space(3))) v4i* lv4i_ptr;

#if defined(__has_builtin)
#if __has_builtin(__builtin_amdgcn_global_store_async_from_lds_b128)
#define FOFE_USE_ASYNC_STORE 1
#endif
#endif

__global__ __launch_bounds__(BLOCK_THREADS) void fofe_encode_kernel(
    const int* __restrict__ x,          // [NUM_WORDS, 32] char codes 0..255
    const float* __restrict__ ff,       // [1] forgetting factor
    float* __restrict__ out)            // [NUM_WORDS, 256]
{
  // 32 KB accumulator (32 words x 256 f32) + 32-entry power table
  __shared__ __align__(16) float acc[WORDS_PER_BLOCK * VOCAB];
  __shared__ float pw[WORD_LEN];

  const int tid  = threadIdx.x;
  const int lane = tid & 31;
  const int wave = tid >> 5;
  const int block_word0 = blockIdx.x * WORDS_PER_BLOCK;

  // Zero the accumulator: 8192 floats / 256 threads = 32 each (coalesced, bank-conflict free)
#pragma unroll
  for (int i = 0; i < (WORDS_PER_BLOCK * VOCAB) / BLOCK_THREADS; ++i)
    acc[tid + i * BLOCK_THREADS] = 0.0f;

  // Power table f^0 .. f^31 (f^0 forced to exactly 1.0 like the reference)
  if (tid < WORD_LEN) {
    float f = ff[0];
    pw[tid] = (tid == 0) ? 1.0f : powf(f, (float)tid);
  }
  __syncthreads();

  // One wave32 per word: lane k holds char position k (WORD_LEN == warpSize == 32).
#pragma unroll
  for (int i = 0; i < WORDS_PER_WAVE; ++i) {
    const int w_in_blk = wave * WORDS_PER_WAVE + i;
    const size_t gidx  = (size_t)(block_word0 + w_in_blk) * WORD_LEN + lane;
    const int c = __builtin_nontemporal_load(&x[gidx]);   // read-once stream

    // #nonzero chars strictly after position `lane`
    const unsigned long long mask = __ballot(c != 0);     // wave32: upper bits zero
    const int s = (int)__popcll(mask >> (lane + 1));      // 64-bit shift: lane+1==32 is safe

    if (c != 0)
      atomicAdd(&acc[w_in_blk * VOCAB + c], pw[s]);       // ds_add_f32; handles duplicate chars
  }
  __syncthreads();

  // Stream the block's 32 KB result to HBM. Region is contiguous:
  // out[block_word0*256 .. +8192). 256 threads x 8 iters x 16 B.
  float* gbase = out + (size_t)block_word0 * VOCAB;

#ifdef FOFE_USE_ASYNC_STORE
  // CDNA5 path: LDS -> global DMA (global_store_async_from_lds_b128),
  // no VGPR round-trip, tracked by ASYNCcnt.
#pragma unroll
  for (int i = 0; i < 8; ++i) {
    const int off = (tid + i * BLOCK_THREADS) * 4;        // float index, 16B aligned
    __builtin_amdgcn_global_store_async_from_lds_b128(
        (gv4i_ptr)(uintptr_t)(gbase + off),
        (lv4i_ptr)(uintptr_t)(const void*)&acc[off],
        /*imm offset*/ 0, /*cpol*/ 0);
  }
#if __has_builtin(__builtin_amdgcn_s_wait_asynccnt)
  __builtin_amdgcn_s_wait_asynccnt(0);
#else
  asm volatile("s_wait_asynccnt 0" ::: "memory");
#endif
#else
  // Fallback: nontemporal 128-bit stores (output is write-once, bypass cache retention)
#pragma unroll
  for (int i = 0; i < 8; ++i) {
    const int off = (tid + i * BLOCK_THREADS) * 4;
    v4f v = *(const v4f*)&acc[off];
    __builtin_nontemporal_store(v, (v4f*)(gbase + off));
  }
#endif
}

extern "C" void kernel_launch(void* const* d_in, const int* in_sizes, int n_in,
                              void* d_out, int out_size, void* d_ws, size_t ws_size,
                              hipStream_t stream) {
  (void)in_sizes; (void)n_in; (void)out_size; (void)d_ws; (void)ws_size;
  const int*   x  = (const int*)d_in[0];     // [NUM_WORDS*32] int
  const float* ff = (const float*)d_in[1];   // [1] float scalar
  float*       out = (float*)d_out;          // [NUM_WORDS*256] float

  const int grid = NUM_WORDS / WORDS_PER_BLOCK;  // 8192 blocks
  fofe_encode_kernel<<<grid, BLOCK_THREADS, 0, stream>>>(x, ff, out);
}